// LightGCN_14766097564263
// MI455X (gfx1250) — compile-verified
//
#include <hip/hip_runtime.h>
#include <hip/hip_bf16.h>

// ---------------- problem constants (match reference) ----------------
#define N_USERS  100000
#define N_ITEMS  50000
#define N_NODES  (N_USERS + N_ITEMS)
#define DIM      64
#define NODE_FLOATS ((size_t)N_NODES * DIM)      // 9,600,000 floats
#define ACC_FLOATS  ((size_t)16384 * DIM)        // per-batch accumulator

typedef __attribute__((ext_vector_type(2))) float v2f;
typedef __attribute__((ext_vector_type(8))) float v8f;

// ---------------------------------------------------------------------
// Kernel 1: concat user_emb and item_emb into E0 (float4 vectorized).
// ---------------------------------------------------------------------
__global__ void lgcn_concat_kernel(const float* __restrict__ ue,
                                   const float* __restrict__ ie,
                                   float* __restrict__ e0) {
    size_t t = (size_t)blockIdx.x * blockDim.x + threadIdx.x;   // float4 index
    const size_t nU4   = (size_t)N_USERS * (DIM / 4);           // 1,600,000
    const size_t nTot4 = (size_t)N_NODES * (DIM / 4);           // 2,400,000
    if (t >= nTot4) return;
    float4 v = (t < nU4) ? ((const float4*)ue)[t]
                         : ((const float4*)ie)[t - nU4];
    ((float4*)e0)[t] = v;
}

// ---------------------------------------------------------------------
// Kernel 2: SpMM scatter:  eout[dst] += val * ein[src]   (atomic f32)
// One wave owns a chunk of 32 contiguous edges (coalesced index loads),
// then broadcasts each edge; 32 lanes move one 64-float row as float2.
// ---------------------------------------------------------------------
__global__ void lgcn_spmm_kernel(const int*   __restrict__ src,
                                 const int*   __restrict__ dst,
                                 const float* __restrict__ val,
                                 const float* __restrict__ ein,
                                 float*       __restrict__ eout,
                                 int nE) {
    const int lane   = threadIdx.x & 31;
    const int wave   = blockIdx.x * (blockDim.x >> 5) + (threadIdx.x >> 5);
    const int nWaves = gridDim.x * (blockDim.x >> 5);

    for (int base = wave * 32; base < nE; base += nWaves * 32) {
        int e = base + lane;
        // prefetch the next chunk of edge metadata into cache
        if (base + nWaves * 32 < nE) {
            __builtin_prefetch(&src[e + nWaves * 32], 0, 0);
            __builtin_prefetch(&dst[e + nWaves * 32], 0, 0);
            __builtin_prefetch(&val[e + nWaves * 32], 0, 0);
        }
        int   s = 0, d = 0;
        float w = 0.0f;
        if (e < nE) { s = src[e]; d = dst[e]; w = val[e]; }
        int cnt = nE - base; if (cnt > 32) cnt = 32;

        for (int j = 0; j < cnt; ++j) {
            int   sj = __shfl(s, j, 32);
            int   dj = __shfl(d, j, 32);
            float wj = __shfl(w, j, 32);
            // 32 lanes x float2 = one 64-float row, fully coalesced (b64 loads)
            float2 x = *(const float2*)(ein + (size_t)sj * DIM + lane * 2);
            float* po = eout + (size_t)dj * DIM + lane * 2;
            unsafeAtomicAdd(po,     wj * x.x);   // global_atomic_add_f32
            unsafeAtomicAdd(po + 1, wj * x.y);
        }
    }
}

// ---------------------------------------------------------------------
// Kernel 3: gather selected rows of emb into per-batch accumulators.
// uacc[b] += emb[users[b]] ; iacc[b] += emb[N_USERS + items[b]]
// ---------------------------------------------------------------------
__global__ void lgcn_accum_kernel(const float* __restrict__ emb,
                                  const int*   __restrict__ users,
                                  const int*   __restrict__ items,
                                  float* __restrict__ uacc,
                                  float* __restrict__ iacc,
                                  int batch) {
    int t    = blockIdx.x * blockDim.x + threadIdx.x; // batch*32 threads
    int b    = t >> 5;
    int lane = t & 31;
    if (b >= batch) return;
    int u  = users[b];
    int it = items[b];
    const float2* ur = (const float2*)(emb + (size_t)u * DIM);
    const float2* ir = (const float2*)(emb + ((size_t)N_USERS + it) * DIM);
    float2* ua = (float2*)(uacc + (size_t)b * DIM);
    float2* ia = (float2*)(iacc + (size_t)b * DIM);
    float2 x = ur[lane], y = ir[lane];
    float2 xa = ua[lane], ya = ia[lane];
    xa.x += x.x; xa.y += x.y;
    ya.x += y.x; ya.y += y.y;
    ua[lane] = xa; ia[lane] = ya;
}

// ---------------------------------------------------------------------
// Kernel 4: batched dot products via V_WMMA_F32_16X16X4_F32.
// One wave handles 16 batch rows: D = U(16x64) * I^T(64x16) accumulated
// over 16 K-chunks of 4 in full f32; gamma = diag(D) / 16
// ( (acc/4)·(acc/4) = acc·acc / 16 ).
//
// A layout (16x4 f32): lanes 0-15 -> M=lane, VGPR{0,1}=K{0,1};
//                      lanes 16-31 -> M=lane-16, VGPR{0,1}=K{2,3}.
// B layout (4x16 f32): same half-lane K split, N = lane&15.
// C/D layout (16x16 f32): VGPR v holds (M=v, N=lane) for lanes 0-15 and
//                         (M=v+8, N=lane-16) for lanes 16-31.
// ---------------------------------------------------------------------
__global__ void lgcn_dot_wmma_kernel(const float* __restrict__ U,
                                     const float* __restrict__ I,
                                     float* __restrict__ out,
                                     int batch) {
    const int lane = threadIdx.x & 31;
    const int wave = blockIdx.x * (blockDim.x >> 5) + (threadIdx.x >> 5);
    const int row0 = wave * 16;
    if (row0 >= batch) return;

    const int m    = lane & 15;          // A row / B column for this lane
    const int koff = (lane >> 4) * 2;    // K sub-offset per lane half
    const float* urow = U + (size_t)(row0 + m) * DIM;
    const float* irow = I + (size_t)(row0 + m) * DIM;

    v8f c = {};
#pragma unroll
    for (int kk = 0; kk < DIM; kk += 4) {
        v2f a = *(const v2f*)(urow + kk + koff);
        v2f b = *(const v2f*)(irow + kk + koff);
        c = __builtin_amdgcn_wmma_f32_16x16x4_f32(
                /*neg_a=*/false, a, /*neg_b=*/false, b,
                /*c_mod=*/(short)0, c, /*reuse_a=*/false, /*reuse_b=*/false);
    }

    // Diagonal extraction per C/D layout:
    //  n = 0..7  lives at lane n      , VGPR n
    //  n = 8..15 lives at lane n + 16 , VGPR n - 8
    const bool lo = (lane < 8);
    const bool hi = (lane >= 24);
    const int idx = lo ? lane : (lane - 24);
    float v = c[0];
#pragma unroll
    for (int i = 1; i < 8; ++i) v = (idx == i) ? c[i] : v;

    const float scale = 1.0f / 16.0f;
    if (lo)      out[row0 + lane]              = v * scale;
    else if (hi) out[row0 + 8 + (lane - 24)]   = v * scale;
}

// ---------------------------------------------------------------------
// Host launch: 3 LightGCN layers (ping-pong SpMM) + WMMA dot readout.
// ---------------------------------------------------------------------
extern "C" void kernel_launch(void* const* d_in, const int* in_sizes, int n_in,
                              void* d_out, int out_size, void* d_ws, size_t ws_size,
                              hipStream_t stream) {
    const int*   users    = (const int*)  d_in[0];
    const int*   items    = (const int*)  d_in[1];
    const int*   edge_src = (const int*)  d_in[2];
    const int*   edge_dst = (const int*)  d_in[3];
    const float* edge_val = (const float*)d_in[4];
    const float* user_emb = (const float*)d_in[5];
    const float* item_emb = (const float*)d_in[6];
    float*       gamma    = (float*)d_out;

    const int batch = in_sizes[0];
    const int nE    = in_sizes[2];

    // Workspace layout (floats): E0 | E1 | UACC | IACC  (~85.2 MB)
    float* E0   = (float*)d_ws;
    float* E1   = E0 + NODE_FLOATS;
    float* UACC = E1 + NODE_FLOATS;
    float* IACC = UACC + (size_t)batch * DIM;

    // Zero the per-batch accumulators (workspace is not re-poisoned to 0).
    hipMemsetAsync(UACC, 0, (size_t)batch * DIM * sizeof(float) * 2, stream);

    // E0 = concat(user_emb, item_emb)
    {
        const int tot4 = N_NODES * (DIM / 4);
        lgcn_concat_kernel<<<(tot4 + 255) / 256, 256, 0, stream>>>(user_emb, item_emb, E0);
    }

    // Layer-0 term of the accumulator: acc rows at the batch indices.
    const int accBlocks = (batch * 32 + 255) / 256;
    lgcn_accum_kernel<<<accBlocks, 256, 0, stream>>>(E0, users, items, UACC, IACC, batch);

    // Three propagation layers with ping-pong buffers.
    float* cur = E0;
    float* nxt = E1;
    const int spmmBlocks = 2048;   // 16384 waves -> ~8 chunks of 32 edges each
    for (int layer = 0; layer < 3; ++layer) {
        hipMemsetAsync(nxt, 0, NODE_FLOATS * sizeof(float), stream);
        lgcn_spmm_kernel<<<spmmBlocks, 256, 0, stream>>>(edge_src, edge_dst, edge_val,
                                                         cur, nxt, nE);
        lgcn_accum_kernel<<<accBlocks, 256, 0, stream>>>(nxt, users, items, UACC, IACC, batch);
        float* t = cur; cur = nxt; nxt = t;
    }

    // gamma[b] = dot(acc_u[b], acc_i[b]) / 16  via WMMA f32 16x16x4.
    const int dotWaves  = batch / 16;                 // 1024 waves
    const int dotBlocks = (dotWaves * 32 + 255) / 256;
    lgcn_dot_wmma_kernel<<<dotBlocks, 256, 0, stream>>>(UACC, IACC, gamma, batch);
}